// GATSyntheticPerturb_49469433316077
// MI455X (gfx1250) — compile-verified
//
#include <hip/hip_runtime.h>
#include <math.h>

#define NN      4096
#define NFEAT   128
#define HID     64
#define NHEADS  4
#define HC0_    64
#define HC1_    32
#define HC2_    32
#define NCLASS_ 10
#define LALPHA  0.2f
#define MAXNNZ  128   // expected row nnz ~17 (incl diag); 128 is >10 sigma safe

typedef __attribute__((ext_vector_type(2))) float v2f;
typedef __attribute__((ext_vector_type(4))) float v4f;
typedef __attribute__((ext_vector_type(8))) float v8f;

// ---------------------------------------------------------------------------
// Pass 1: one wave32 per row. Stream sub_adj row with non-temporal b128 loads
// (512 B per wave per instruction -> 4x fewer VMEM ops on the 64MB scan that
// sets the bandwidth floor). Compact nonzeros into ELL via per-lane count +
// shfl_up inclusive scan; fused degree d = 1 + sum(sigmoid(P_symm)).
// ---------------------------------------------------------------------------
__global__ void k_build_ell(const float* __restrict__ adj,
                            const float* __restrict__ P,
                            int* __restrict__ cols, float* __restrict__ vals,
                            int* __restrict__ nnz, float* __restrict__ deg) {
  int wid  = (blockIdx.x * blockDim.x + threadIdx.x) >> 5;
  int lane = threadIdx.x & 31;
  if (wid >= NN) return;
  const int i = wid;
  const v4f* arow4 = (const v4f*)(adj + (size_t)i * NN);
  const int base = i * MAXNNZ;
  int count = 0;
  float dsum = 0.f;
  for (int j0 = 0; j0 < NN; j0 += 128) {           // 128 cols per wave step
    v4f av = __builtin_nontemporal_load(arow4 + (j0 >> 2) + lane);
    int cnt = (av.x != 0.f) + (av.y != 0.f) + (av.z != 0.f) + (av.w != 0.f);
    int inc = cnt;                                  // inclusive scan over lanes
#pragma unroll
    for (int d = 1; d < 32; d <<= 1) {
      int t = __shfl_up(inc, d, 32);
      if (lane >= d) inc += t;
    }
    int total = __shfl(inc, 31, 32);
    int off = count + inc - cnt;                    // exclusive offset
    float vv[4] = {av.x, av.y, av.z, av.w};
#pragma unroll
    for (int e = 0; e < 4; ++e) {
      if (vv[e] != 0.f) {
        int j = j0 + lane * 4 + e;
        // lower-tri packed index: (i>=j) ? i*(i+1)/2+j : j*(j+1)/2+i
        int pidx = (i >= j) ? (i * (i + 1) / 2 + j) : (j * (j + 1) / 2 + i);
        float a = 1.f / (1.f + __expf(-P[pidx]));   // sigmoid * adj(==1)
        if (off < MAXNNZ - 1) { cols[base + off] = j; vals[base + off] = a; }
        ++off;
        dsum += a;
      }
    }
    count += total;
  }
  for (int off = 16; off > 0; off >>= 1) dsum += __shfl_xor(dsum, off, 32);
  if (count > MAXNNZ - 1) count = MAXNNZ - 1;
  if (lane == 0) {
    cols[base + count] = i;        // + identity
    vals[base + count] = 1.f;
    nnz[i] = count + 1;
    deg[i] = dsum + 1.f;
  }
}

__global__ void k_dinv(const float* __restrict__ deg, float* __restrict__ dinv) {
  int i = blockIdx.x * blockDim.x + threadIdx.x;
  if (i >= NN) return;
  float d = deg[i];
  dinv[i] = (d > 0.f) ? rsqrtf(d) : 0.f;
}

// vals <- dinv[i] * a * dinv[j]  (norm_adj entries; operates only on nnz)
__global__ void k_scale_vals(const int* __restrict__ cols, float* __restrict__ vals,
                             const int* __restrict__ nnz, const float* __restrict__ dinv) {
  int wid  = (blockIdx.x * blockDim.x + threadIdx.x) >> 5;
  int lane = threadIdx.x & 31;
  if (wid >= NN) return;
  const int i = wid, base = i * MAXNNZ, n = nnz[i];
  float di = dinv[i];
  for (int k = lane; k < n; k += 32)
    vals[base + k] = di * vals[base + k] * dinv[cols[base + k]];
}

// ---------------------------------------------------------------------------
// f32 WMMA GEMM, C[M x N] = A[M x K] @ B[K x N], V_WMMA_F32_16X16X4_F32.
// Fully compile-time shapes (K, LDA, LDB, LDC) so all inner-loop addressing
// folds into immediate offsets. One wave owns 16 rows x NT*16 cols: the A
// fragment is loaded ONCE per k-step and reused across NT accumulators; all
// NT B fragments are fetched before the WMMA chain so loads clause together
// and retire under staggered s_wait_loadcnt while WMMAs issue.
// A frag (16x4): lanes 0-15 row=lane K={0,1}; lanes 16-31 row=lane-16 K={2,3}.
// B frag (4x16): same K split, N = lane%16. C/D: VGPR r = row r + 8*(lane/16).
// gridDim.z = head index; B/C advance by the given per-head strides.
// ---------------------------------------------------------------------------
template <int NT, int K, int LDA, int LDB, int LDC>
__global__ void k_wmma_gemm(const float* __restrict__ A,
                            const float* __restrict__ B,
                            float* __restrict__ C,
                            int strideB, int strideC) {
  B += (size_t)blockIdx.z * strideB;
  C += (size_t)blockIdx.z * strideC;
  const int n0   = blockIdx.x * (16 * NT);
  const int m0   = blockIdx.y * 16;
  const int lane = threadIdx.x & 31;
  const int half = lane >> 4;
  const int lm   = lane & 15;
  v8f acc[NT];
#pragma unroll
  for (int nt = 0; nt < NT; ++nt) acc[nt] = (v8f){};
  const float* arow = A + (size_t)(m0 + lm) * LDA + 2 * half;
  const float* bcol = B + (size_t)(2 * half) * LDB + n0 + lm;
#pragma unroll 4
  for (int k0 = 0; k0 < K; k0 += 4) {
    v2f a;
    a.x = arow[k0 + 0];
    a.y = arow[k0 + 1];
    v2f b[NT];
#pragma unroll
    for (int nt = 0; nt < NT; ++nt) {       // clause all B loads first
      b[nt].x = bcol[(size_t)k0 * LDB + nt * 16];
      b[nt].y = bcol[(size_t)(k0 + 1) * LDB + nt * 16];
    }
#pragma unroll
    for (int nt = 0; nt < NT; ++nt)         // then the WMMA chain
      acc[nt] = __builtin_amdgcn_wmma_f32_16x16x4_f32(false, a, false, b[nt],
                                                      (short)0, acc[nt],
                                                      false, false);
  }
  float* crow = C + (size_t)(m0 + 8 * half) * LDC + n0 + lm;
#pragma unroll
  for (int nt = 0; nt < NT; ++nt)
#pragma unroll
    for (int r = 0; r < 8; ++r) crow[(size_t)r * LDC + nt * 16] = acc[nt][r];
}

// s1[h*N+i] = Wh[i,h*64:]·a1_h ; s2 likewise with a2_h
__global__ void k_scores(const float* __restrict__ Wh, int ldw,
                         const float* __restrict__ a, int heads,
                         float* __restrict__ s1, float* __restrict__ s2) {
  int tid = blockIdx.x * blockDim.x + threadIdx.x;
  if (tid >= NN * heads) return;
  int h = tid / NN, i = tid % NN;
  const float* w  = Wh + (size_t)i * ldw + h * HID;
  const float* a1 = a + h * (2 * HID);
  const float* a2 = a1 + HID;
  float r1 = 0.f, r2 = 0.f;
  for (int d = 0; d < HID; ++d) { float v = w[d]; r1 += v * a1[d]; r2 += v * a2[d]; }
  s1[tid] = r1;
  s2[tid] = r2;
}

// Sparse masked softmax attention + aggregation + ELU.
// One wave per (row, head=blockIdx.y); lanes cover 64 output dims (2 each).
// Masked-out entries contribute exp(-9e15 - m) == 0, so only nnz matter.
__global__ void k_attn(const int* __restrict__ cols, const int* __restrict__ nnz,
                       const float* __restrict__ Wh, int ldw,
                       const float* __restrict__ s1, const float* __restrict__ s2,
                       float* __restrict__ out, int ldo) {
  int wid  = (blockIdx.x * blockDim.x + threadIdx.x) >> 5;
  int lane = threadIdx.x & 31;
  int h    = blockIdx.y;
  if (wid >= NN) return;
  const int i = wid, base = i * MAXNNZ, n = nnz[i];
  const float s1v = s1[h * NN + i];
  const int* ci = cols + base;
  float m = -INFINITY, den = 0.f;               // online softmax
  for (int k = 0; k < n; ++k) {
    float e = s1v + s2[h * NN + ci[k]];
    e = (e > 0.f) ? e : LALPHA * e;             // leaky_relu
    float mn = fmaxf(m, e);
    den = den * __expf(m - mn) + __expf(e - mn);
    m = mn;
  }
  float inv = 1.f / den;
  float acc0 = 0.f, acc1 = 0.f;
  for (int k = 0; k < n; ++k) {
    int j = ci[k];
    float e = s1v + s2[h * NN + j];
    e = (e > 0.f) ? e : LALPHA * e;
    float w = __expf(e - m) * inv;
    const float* wr = Wh + (size_t)j * ldw + h * HID;
    acc0 += w * wr[lane];
    acc1 += w * wr[lane + 32];
  }
  float* o = out + (size_t)i * ldo + h * HID;
  o[lane]      = (acc0 > 0.f) ? acc0 : __expf(acc0) - 1.f;   // elu
  o[lane + 32] = (acc1 > 0.f) ? acc1 : __expf(acc1) - 1.f;
}

// out[N x 32] = norm_adj @ Y + bias ; one wave per row, lane = column
__global__ void k_spmm(const int* __restrict__ cols, const float* __restrict__ vals,
                       const int* __restrict__ nnz, const float* __restrict__ Y,
                       const float* __restrict__ bias, float* __restrict__ out) {
  int wid  = (blockIdx.x * blockDim.x + threadIdx.x) >> 5;
  int lane = threadIdx.x & 31;
  if (wid >= NN) return;
  const int i = wid, base = i * MAXNNZ, n = nnz[i];
  float acc = bias[lane];
  for (int k = 0; k < n; ++k)
    acc += vals[base + k] * Y[(size_t)cols[base + k] * 32 + lane];
  out[(size_t)i * 32 + lane] = acc;
}

// logits = [x1,x2] @ W_lin + b_lin ; log_softmax over 10 classes. Thread/row.
__global__ void k_logits(const float* __restrict__ x1, const float* __restrict__ x2,
                         const float* __restrict__ Wl, const float* __restrict__ bl,
                         float* __restrict__ out) {
  int i = blockIdx.x * blockDim.x + threadIdx.x;
  if (i >= NN) return;
  float l[NCLASS_];
#pragma unroll
  for (int c = 0; c < NCLASS_; ++c) l[c] = bl[c];
  for (int k = 0; k < 32; ++k) {
    float v = x1[(size_t)i * 32 + k];
#pragma unroll
    for (int c = 0; c < NCLASS_; ++c) l[c] += v * Wl[k * NCLASS_ + c];
  }
  for (int k = 0; k < 32; ++k) {
    float v = x2[(size_t)i * 32 + k];
#pragma unroll
    for (int c = 0; c < NCLASS_; ++c) l[c] += v * Wl[(32 + k) * NCLASS_ + c];
  }
  float m = l[0];
#pragma unroll
  for (int c = 1; c < NCLASS_; ++c) m = fmaxf(m, l[c]);
  float s = 0.f;
#pragma unroll
  for (int c = 0; c < NCLASS_; ++c) s += __expf(l[c] - m);
  float lse = m + __logf(s);
#pragma unroll
  for (int c = 0; c < NCLASS_; ++c) out[(size_t)i * NCLASS_ + c] = l[c] - lse;
}

static inline size_t alignup(size_t x) { return (x + 255) & ~(size_t)255; }

extern "C" void kernel_launch(void* const* d_in, const int* in_sizes, int n_in,
                              void* d_out, int out_size, void* d_ws, size_t ws_size,
                              hipStream_t stream) {
  const float* x       = (const float*)d_in[0];   // [4096,128]
  const float* sub_adj = (const float*)d_in[1];   // [4096,4096]
  const float* P_vec   = (const float*)d_in[2];   // [N*(N+1)/2]
  const float* W_att   = (const float*)d_in[3];   // [4,128,64]
  const float* a_att   = (const float*)d_in[4];   // [4,128,1]
  const float* W_out   = (const float*)d_in[5];   // [256,64]
  const float* a_out   = (const float*)d_in[6];   // [128,1]
  const float* W_c1    = (const float*)d_in[7];   // [64,32]
  const float* b_c1    = (const float*)d_in[8];   // [32]
  const float* W_c2    = (const float*)d_in[9];   // [32,32]
  const float* b_c2    = (const float*)d_in[10];  // [32]
  const float* W_lin   = (const float*)d_in[11];  // [64,10]
  const float* b_lin   = (const float*)d_in[12];  // [10]
  float* out = (float*)d_out;                     // [4096,10]

  char* ws = (char*)d_ws;
  size_t off = 0;
  auto alloc = [&](size_t bytes) { char* p = ws + off; off = alignup(off + bytes); return p; };
  float* deg    = (float*)alloc(NN * 4);
  float* dinv   = (float*)alloc(NN * 4);
  int*   nnz    = (int*)  alloc(NN * 4);
  int*   cols   = (int*)  alloc((size_t)NN * MAXNNZ * 4);
  float* vals   = (float*)alloc((size_t)NN * MAXNNZ * 4);
  float* Wh_all = (float*)alloc((size_t)NN * 256 * 4);   // pre-activation per-head Wh
  float* s1a    = (float*)alloc((size_t)NHEADS * NN * 4);
  float* s2a    = (float*)alloc((size_t)NHEADS * NN * 4);
  float* xh     = (float*)alloc((size_t)NN * 256 * 4);   // elu(att@Wh) concat
  float* Wh2    = (float*)alloc((size_t)NN * 64 * 4);
  float* s1b    = (float*)alloc(NN * 4);
  float* s2b    = (float*)alloc(NN * 4);
  float* xo     = (float*)alloc((size_t)NN * 64 * 4);
  float* t1     = (float*)alloc((size_t)NN * 32 * 4);
  float* x1     = (float*)alloc((size_t)NN * 32 * 4);
  float* t2     = (float*)alloc((size_t)NN * 32 * 4);
  float* x2     = (float*)alloc((size_t)NN * 32 * 4);
  (void)ws_size; (void)n_in; (void)in_sizes; (void)out_size;

  const int WPB = 8;               // wave32s per 256-thread block
  dim3 blk(256);
  dim3 rowGrid(NN / WPB);          // one wave per row

  // 1) single streaming NT b128 pass over adjacency: ELL structure + degrees
  k_build_ell<<<rowGrid, blk, 0, stream>>>(sub_adj, P_vec, cols, vals, nnz, deg);
  // 2) d^-1/2
  k_dinv<<<NN / 256, blk, 0, stream>>>(deg, dinv);
  // 3) norm_adj values
  k_scale_vals<<<rowGrid, blk, 0, stream>>>(cols, vals, nnz, dinv);

  // 4) all-head Wh = x @ W_att[h] (WMMA f32, K=128), fused over gridDim.z
  k_wmma_gemm<4, NFEAT, NFEAT, HID, 256>
      <<<dim3(1, NN / 16, NHEADS), dim3(32), 0, stream>>>(
          x, W_att, Wh_all, NFEAT * HID /*B head stride*/, HID /*C head stride*/);
  // 5) attention score vectors
  k_scores<<<(NN * NHEADS) / 256, blk, 0, stream>>>(Wh_all, 256, a_att, NHEADS, s1a, s2a);
  // 6) sparse softmax attention + ELU -> xh [N,256]
  k_attn<<<dim3(NN / WPB, NHEADS), blk, 0, stream>>>(cols, nnz, Wh_all, 256, s1a, s2a, xh, 256);

  // 7) layer-2: Wh2 = xh @ W_out (K=256)
  k_wmma_gemm<2, 256, 256, HID, HID>
      <<<dim3(2, NN / 16, 1), dim3(32), 0, stream>>>(xh, W_out, Wh2, 0, 0);
  k_scores<<<NN / 256, blk, 0, stream>>>(Wh2, HID, a_out, 1, s1b, s2b);
  k_attn<<<dim3(NN / WPB, 1), blk, 0, stream>>>(cols, nnz, Wh2, HID, s1b, s2b, xo, HID);

  // 8) conv1: x1 = norm_adj @ (xo @ W_c1) + b_c1
  k_wmma_gemm<2, HID, HID, HC1_, HC1_>
      <<<dim3(1, NN / 16, 1), dim3(32), 0, stream>>>(xo, W_c1, t1, 0, 0);
  k_spmm<<<rowGrid, blk, 0, stream>>>(cols, vals, nnz, t1, b_c1, x1);
  //    conv2: x2 = norm_adj @ (x1 @ W_c2) + b_c2
  k_wmma_gemm<2, HC1_, HC1_, HC2_, HC2_>
      <<<dim3(1, NN / 16, 1), dim3(32), 0, stream>>>(x1, W_c2, t2, 0, 0);
  k_spmm<<<rowGrid, blk, 0, stream>>>(cols, vals, nnz, t2, b_c2, x2);

  // 9) classifier + log_softmax
  k_logits<<<NN / 256, blk, 0, stream>>>(x1, x2, W_lin, b_lin, out);
}